// GraphMultisetTransformerGNN_52845277610105
// MI455X (gfx1250) — compile-verified
//
#include <hip/hip_runtime.h>
#include <math.h>

#define NN 100000
#define NE 1600000

typedef __attribute__((ext_vector_type(2))) float v2f;
typedef __attribute__((ext_vector_type(8))) float v8f;

// ---------------------------------------------------------------------------
// Generic WMMA GEMM: C[M,N] = act(alpha * A[M,128] @ B + bias)
// BTRANS==0: B row-major [128,N]   (element (k,j) at B[k*N+j])
// BTRANS==1: B row-major [N,128], used transposed (element (k,j) at B[j*128+k])
// act: 0=none, 1=relu, 2=sigmoid
// One wave per 16x(16*NTILE) output strip; K=128 via chained
// V_WMMA_F32_16X16X4_F32. A fragment reused across NTILE column tiles.
// Fragments are double-buffered: the k+4 loads issue before the k-step WMMAs
// so memory latency overlaps the matrix pipe instead of stalling it.
// A row index is clamped (not predicated): out-of-range rows only feed
// accumulator rows that are never stored, keeping loads straight-line.
// ---------------------------------------------------------------------------
template <int BTRANS, int NTILE>
__global__ __launch_bounds__(32)
void k_gemm128(const float* __restrict__ A, const float* __restrict__ B,
               float* __restrict__ C, int M, int N,
               float alpha, const float* __restrict__ bias, int act)
{
    const int tm   = blockIdx.x * 16;
    const int tn0  = blockIdx.y * 16 * NTILE;
    const int lane = threadIdx.x;
    const int half = lane >> 4;   // selects K slot pair per ISA A/B layout
    const int l    = lane & 15;

    v8f acc[NTILE];
#pragma unroll
    for (int tt = 0; tt < NTILE; ++tt)
        acc[tt] = (v8f){0.f, 0.f, 0.f, 0.f, 0.f, 0.f, 0.f, 0.f};

    const int arow  = min(tm + l, M - 1);     // clamp instead of predicate
    const float* Ar = A + (size_t)arow * 128 + 2 * half;

    // per-tile B base pointers (K advances by stride_k per k-step of 4)
    const float* Bp[NTILE];
#pragma unroll
    for (int tt = 0; tt < NTILE; ++tt) {
        const int bcol = tn0 + tt * 16 + l;
        Bp[tt] = BTRANS ? (B + (size_t)bcol * 128 + 2 * half)
                        : (B + (size_t)(2 * half) * N + bcol);
    }

    auto loadB = [&](int tt, int kk) -> v2f {
        v2f b;
        if (BTRANS) {
            b = *(const v2f*)(Bp[tt] + kk);
        } else {
            b.x = Bp[tt][(size_t)kk * N];
            b.y = Bp[tt][(size_t)(kk + 1) * N];
        }
        return b;
    };

    // prologue: fragments for k0 = 0
    v2f a_cur = *(const v2f*)(Ar);
    v2f b_cur[NTILE];
#pragma unroll
    for (int tt = 0; tt < NTILE; ++tt) b_cur[tt] = loadB(tt, 0);

#pragma unroll
    for (int k0 = 0; k0 < 128; k0 += 4) {
        v2f a_nxt;
        v2f b_nxt[NTILE];
        if (k0 + 4 < 128) {
            // prefetch the next k-step's fragments before consuming this one
            a_nxt = *(const v2f*)(Ar + k0 + 4);
#pragma unroll
            for (int tt = 0; tt < NTILE; ++tt) b_nxt[tt] = loadB(tt, k0 + 4);
        }
#pragma unroll
        for (int tt = 0; tt < NTILE; ++tt)
            acc[tt] = __builtin_amdgcn_wmma_f32_16x16x4_f32(
                false, a_cur, false, b_cur[tt], (short)0, acc[tt], false, false);
        if (k0 + 4 < 128) {
            a_cur = a_nxt;
#pragma unroll
            for (int tt = 0; tt < NTILE; ++tt) b_cur[tt] = b_nxt[tt];
        }
    }

    // C/D layout: lanes 0-15 N=l, M=tm+r ; lanes 16-31 N=l, M=tm+8+r
    const int rbase = tm + half * 8;
#pragma unroll
    for (int tt = 0; tt < NTILE; ++tt) {
        const int bcol = tn0 + tt * 16 + l;
        const float bv = bias ? bias[bcol] : 0.f;
#pragma unroll
        for (int r = 0; r < 8; ++r) {
            const int row = rbase + r;
            if (row < M) {
                float v = acc[tt][r] * alpha + bv;
                if (act == 1)      v = fmaxf(v, 0.f);
                else if (act == 2) v = 1.f / (1.f + __expf(-v));
                C[(size_t)row * N + bcol] = v;
            }
        }
    }
}

// ---------------------------------------------------------------------------
// Small utility kernels
// ---------------------------------------------------------------------------
__global__ void k_fill(float* __restrict__ p, float v, int n)
{
    int t = blockIdx.x * blockDim.x + threadIdx.x;
    if (t < n) p[t] = v;
}

__global__ void k_deg(const int* __restrict__ dst, float* __restrict__ deg, int nE)
{
    int t = blockIdx.x * blockDim.x + threadIdx.x;
    if (t < nE) atomicAdd(&deg[dst[t]], 1.0f);
}

// dinv = (indeg + 1 self-loop)^-1/2, in place
__global__ void k_dinv(float* __restrict__ d, int n)
{
    int t = blockIdx.x * blockDim.x + threadIdx.x;
    if (t < n) d[t] = rsqrtf(d[t] + 1.0f);
}

// Self-loop term + bias:  out[i,c] = xw[i,c]*dinv[i]^2 + bias[c]
__global__ void k_selfbias(const float* __restrict__ xw, const float* __restrict__ dinv,
                           const float* __restrict__ bias, float* __restrict__ out, int n)
{
    size_t t = (size_t)blockIdx.x * blockDim.x + threadIdx.x;
    if (t >= (size_t)n * 128) return;
    int i = (int)(t >> 7), c = (int)(t & 127);
    float di = dinv[i];
    out[t] = xw[t] * di * di + bias[c];
}

// Edge scatter: one wave per edge, float4 per lane, atomic f32 adds (resolve in L2)
__global__ __launch_bounds__(256)
void k_scatter(const float* __restrict__ xw, const float* __restrict__ dinv,
               const int* __restrict__ src, const int* __restrict__ dst,
               float* __restrict__ out, int nE)
{
    size_t t = (size_t)blockIdx.x * blockDim.x + threadIdx.x;
    int e = (int)(t >> 5), lane = (int)(t & 31);
    if (e >= nE) return;
    int s = src[e], d = dst[e];
    float w = dinv[s] * dinv[d];
    float4 v = ((const float4*)(xw + (size_t)s * 128))[lane];
    float* o = out + (size_t)d * 128 + lane * 4;
    atomicAdd(o + 0, v.x * w);
    atomicAdd(o + 1, v.y * w);
    atomicAdd(o + 2, v.z * w);
    atomicAdd(o + 3, v.w * w);
}

__global__ void k_relu(float* __restrict__ p, size_t n)
{
    size_t t = (size_t)blockIdx.x * blockDim.x + threadIdx.x;
    if (t < n) p[t] = fmaxf(p[t], 0.f);
}

// Repack per-head weight [8,128,16] (h,e,d) -> combined row-major B[128,128] (e, h*16+d)
__global__ void k_repack(const float* __restrict__ W, float* __restrict__ B)
{
    int t = blockIdx.x * blockDim.x + threadIdx.x;  // 16384
    int e = t >> 7, c = t & 127;
    int h = c >> 4, d = c & 15;
    B[t] = W[h * 2048 + e * 16 + d];
}

// ---------------------------------------------------------------------------
// Streaming softmax attention pooling over all nodes.
// One block per (head, seed). Per-thread running (m, l, acc[16]) flash-merge,
// then log-sum-exp tree reduction in LDS.
// ---------------------------------------------------------------------------
__global__ __launch_bounds__(256)
void k_attnpool(const float* __restrict__ K, const float* __restrict__ V,
                const float* __restrict__ Q, float* __restrict__ O, int n)
{
    const int h = blockIdx.x & 7;
    const int s = blockIdx.x >> 3;
    const int t = threadIdx.x;

    float q[16];
#pragma unroll
    for (int d = 0; d < 16; ++d) q[d] = Q[s * 128 + h * 16 + d];

    float m = -1e30f, l = 0.f;
    float acc[16];
#pragma unroll
    for (int d = 0; d < 16; ++d) acc[d] = 0.f;

    for (int i = t; i < n; i += 256) {
        const float* kr = K + (size_t)i * 128 + h * 16;
        float sc = 0.f;
#pragma unroll
        for (int d = 0; d < 16; ++d) sc += q[d] * kr[d];
        float mn = fmaxf(m, sc);
        float e0 = __expf(m - mn);
        float p  = __expf(sc - mn);
        l = l * e0 + p;
        const float* vr = V + (size_t)i * 128 + h * 16;
#pragma unroll
        for (int d = 0; d < 16; ++d) acc[d] = acc[d] * e0 + p * vr[d];
        m = mn;
    }

    __shared__ float sm[256], sl[256], sac[256 * 16];
    sm[t] = m; sl[t] = l;
#pragma unroll
    for (int d = 0; d < 16; ++d) sac[t * 16 + d] = acc[d];
    __syncthreads();
    for (int off = 128; off > 0; off >>= 1) {
        if (t < off) {
            float m1 = sm[t], m2 = sm[t + off];
            float mn = fmaxf(m1, m2);
            float e1 = __expf(m1 - mn), e2 = __expf(m2 - mn);
            sm[t] = mn;
            sl[t] = sl[t] * e1 + sl[t + off] * e2;
            for (int d = 0; d < 16; ++d)
                sac[t * 16 + d] = sac[t * 16 + d] * e1 + sac[(t + off) * 16 + d] * e2;
        }
        __syncthreads();
    }
    if (t < 16) O[s * 128 + h * 16 + t] = sac[t] / sl[0];
}

// Self-attention over 32 seeds: one block, thread = (head, seed)
__global__ __launch_bounds__(256)
void k_selfattn32(const float* __restrict__ qkv, float* __restrict__ O)
{
    int t = threadIdx.x;
    int h = t >> 5;   // 0..7
    int i = t & 31;   // seed
    float q[16];
#pragma unroll
    for (int d = 0; d < 16; ++d) q[d] = qkv[i * 384 + h * 16 + d] * 0.25f;
    float sc[32];
    float mx = -1e30f;
#pragma unroll
    for (int j = 0; j < 32; ++j) {
        float s = 0.f;
        const float* kr = qkv + j * 384 + 128 + h * 16;
#pragma unroll
        for (int d = 0; d < 16; ++d) s += q[d] * kr[d];
        sc[j] = s; mx = fmaxf(mx, s);
    }
    float l = 0.f;
#pragma unroll
    for (int j = 0; j < 32; ++j) { sc[j] = __expf(sc[j] - mx); l += sc[j]; }
    float acc[16];
#pragma unroll
    for (int d = 0; d < 16; ++d) acc[d] = 0.f;
#pragma unroll
    for (int j = 0; j < 32; ++j) {
        const float* vr = qkv + j * 384 + 256 + h * 16;
        float p = sc[j];
#pragma unroll
        for (int d = 0; d < 16; ++d) acc[d] += p * vr[d];
    }
#pragma unroll
    for (int d = 0; d < 16; ++d) O[i * 128 + h * 16 + d] = acc[d] / l;
}

// a3 attention: 1 seed, 8 heads, 32 "nodes"; one block of 128 (thread = (h,d))
__global__ __launch_bounds__(128)
void k_attn3(const float* __restrict__ K, const float* __restrict__ V,
             const float* __restrict__ q, float* __restrict__ o)
{
    int t = threadIdx.x;
    int h = t >> 4, d = t & 15;
    float qh[16];
#pragma unroll
    for (int dd = 0; dd < 16; ++dd) qh[dd] = q[h * 16 + dd];
    float sc[32];
    float mx = -1e30f;
#pragma unroll
    for (int j = 0; j < 32; ++j) {
        float s = 0.f;
#pragma unroll
        for (int dd = 0; dd < 16; ++dd) s += qh[dd] * K[j * 128 + h * 16 + dd];
        sc[j] = s; mx = fmaxf(mx, s);
    }
    float l = 0.f;
#pragma unroll
    for (int j = 0; j < 32; ++j) { sc[j] = __expf(sc[j] - mx); l += sc[j]; }
    float a = 0.f;
#pragma unroll
    for (int j = 0; j < 32; ++j) a += sc[j] * V[j * 128 + h * 16 + d];
    o[h * 16 + d] = a / l;
}

// Z = LayerNorm(X + Y) * w + b over rows of width 128; one block (128 thr) per row
__global__ __launch_bounds__(128)
void k_lnadd(const float* __restrict__ X, const float* __restrict__ Y,
             const float* __restrict__ w, const float* __restrict__ b,
             float* __restrict__ Z)
{
    int r = blockIdx.x, t = threadIdx.x;
    float v = X[r * 128 + t] + Y[r * 128 + t];
    __shared__ float red[128];
    red[t] = v; __syncthreads();
    for (int off = 64; off > 0; off >>= 1) { if (t < off) red[t] += red[t + off]; __syncthreads(); }
    float mean = red[0] * (1.f / 128.f);
    __syncthreads();
    float dv = v - mean;
    red[t] = dv * dv; __syncthreads();
    for (int off = 64; off > 0; off >>= 1) { if (t < off) red[t] += red[t + off]; __syncthreads(); }
    float var = red[0] * (1.f / 128.f);
    Z[r * 128 + t] = dv * rsqrtf(var + 1e-5f) * w[t] + b[t];
}

// ---------------------------------------------------------------------------
static inline void launch_gemm(const float* A, const float* B, float* C,
                               int M, int N, int btrans, float alpha,
                               const float* bias, int act, hipStream_t stream)
{
    const int mg = (M + 15) / 16;
    if (N % 64 == 0) {
        dim3 g(mg, N / 64);
        if (btrans) k_gemm128<1, 4><<<g, 32, 0, stream>>>(A, B, C, M, N, alpha, bias, act);
        else        k_gemm128<0, 4><<<g, 32, 0, stream>>>(A, B, C, M, N, alpha, bias, act);
    } else {                       // N == 32
        dim3 g(mg, N / 32);
        if (btrans) k_gemm128<1, 2><<<g, 32, 0, stream>>>(A, B, C, M, N, alpha, bias, act);
        else        k_gemm128<0, 2><<<g, 32, 0, stream>>>(A, B, C, M, N, alpha, bias, act);
    }
}

extern "C" void kernel_launch(void* const* d_in, const int* in_sizes, int n_in,
                              void* d_out, int out_size, void* d_ws, size_t ws_size,
                              hipStream_t stream)
{
    (void)in_sizes; (void)n_in; (void)out_size; (void)ws_size;

    // setup_inputs() dict insertion order (recursive)
    const float* x     = (const float*)d_in[0];
    const float* Wc[3] = {(const float*)d_in[1], (const float*)d_in[3], (const float*)d_in[5]};
    const float* bc[3] = {(const float*)d_in[2], (const float*)d_in[4], (const float*)d_in[6]};
    // a1 (attn_p): seed, Wk, bk, Wv, bv, q, Wo, bo, lnh_w, lnh_b, lnz_w, lnz_b, Wff, bff
    const float* seed1 = (const float*)d_in[7];
    const float* Wk1   = (const float*)d_in[8];
    const float* bk1   = (const float*)d_in[9];
    const float* Wv1   = (const float*)d_in[10];
    const float* bv1   = (const float*)d_in[11];
    const float* q1w   = (const float*)d_in[12];
    const float* Wo1   = (const float*)d_in[13];
    const float* bo1   = (const float*)d_in[14];
    const float* lnh1w = (const float*)d_in[15];
    const float* lnh1b = (const float*)d_in[16];
    const float* lnz1w = (const float*)d_in[17];
    const float* lnz1b = (const float*)d_in[18];
    const float* Wff1  = (const float*)d_in[19];
    const float* bff1  = (const float*)d_in[20];
    // a2: Wi, bi, Wo, bo, lnh_w, lnh_b, lnz_w, lnz_b, Wff, bff
    const float* Wi2   = (const float*)d_in[21];
    const float* bi2   = (const float*)d_in[22];
    const float* Wo2   = (const float*)d_in[23];
    const float* bo2   = (const float*)d_in[24];
    const float* lnh2w = (const float*)d_in[25];
    const float* lnh2b = (const float*)d_in[26];
    const float* lnz2w = (const float*)d_in[27];
    const float* lnz2b = (const float*)d_in[28];
    const float* Wff2  = (const float*)d_in[29];
    const float* bff2  = (const float*)d_in[30];
    // a3
    const float* seed3 = (const float*)d_in[31];
    const float* Wk3   = (const float*)d_in[32];
    const float* bk3   = (const float*)d_in[33];
    const float* Wv3   = (const float*)d_in[34];
    const float* bv3   = (const float*)d_in[35];
    const float* q3w   = (const float*)d_in[36];
    const float* Wo3   = (const float*)d_in[37];
    const float* bo3   = (const float*)d_in[38];
    const float* lnh3w = (const float*)d_in[39];
    const float* lnh3b = (const float*)d_in[40];
    const float* lnz3w = (const float*)d_in[41];
    const float* lnz3b = (const float*)d_in[42];
    const float* Wff3  = (const float*)d_in[43];
    const float* bff3  = (const float*)d_in[44];
    const float* headW = (const float*)d_in[45];
    const float* headb = (const float*)d_in[46];
    const int*   ei    = (const int*)d_in[47];
    const int* src = ei;
    const int* dst = ei + NE;

    float* ws = (float*)d_ws;

    // --- workspace layout (floats) ---
    const size_t NB = (size_t)NN * 128;
    float* buf0 = ws;
    float* buf1 = ws + NB;
    float* buf2 = ws + 2 * NB;
    size_t off  = 3 * NB;
    float* dinv = ws + off;            off += NN;
    float* Bk   = ws + off;            off += 16384;
    float* Bv   = ws + off;            off += 16384;
    float* wq1  = ws + off;            off += 4096;
    float* wo1  = ws + off;            off += 4096;
    float* wp1  = ws + off;            off += 4096;
    float* wh1  = ws + off;            off += 4096;
    float* wf1  = ws + off;            off += 4096;
    float* wz1  = ws + off;            off += 4096;
    float* wqkv = ws + off;            off += 12288;
    float* wo2  = ws + off;            off += 4096;
    float* wp2  = ws + off;            off += 4096;
    float* wh2  = ws + off;            off += 4096;
    float* wf2  = ws + off;            off += 4096;
    float* wz2  = ws + off;            off += 4096;
    float* wk3  = ws + off;            off += 4096;
    float* wv3  = ws + off;            off += 4096;
    float* wq3  = ws + off;            off += 128;
    float* wo3  = ws + off;            off += 128;
    float* wp3  = ws + off;            off += 128;
    float* wh3  = ws + off;            off += 128;
    float* wf3  = ws + off;            off += 128;

    const int gN   = (NN + 255) / 256;          // per-node grid
    const int gNC  = (int)((NB + 255) / 256);   // per-element grid
    const int gE   = (NE + 255) / 256;          // per-edge grid
    const int gE32 = (int)(((size_t)NE * 32 + 255) / 256);

    // Degree normalization: dinv = (indeg + 1)^-1/2
    k_fill<<<gN, 256, 0, stream>>>(dinv, 0.f, NN);
    k_deg<<<gE, 256, 0, stream>>>(dst, dinv, NE);
    k_dinv<<<gN, 256, 0, stream>>>(dinv, NN);

    // --- 3x GCN + ReLU ---
    // conv1: x -> buf1
    launch_gemm(x, Wc[0], buf0, NN, 128, 0, 1.f, nullptr, 0, stream);
    k_selfbias<<<gNC, 256, 0, stream>>>(buf0, dinv, bc[0], buf1, NN);
    k_scatter<<<gE32, 256, 0, stream>>>(buf0, dinv, src, dst, buf1, NE);
    k_relu<<<gNC, 256, 0, stream>>>(buf1, NB);
    // conv2: buf1 -> buf2
    launch_gemm(buf1, Wc[1], buf0, NN, 128, 0, 1.f, nullptr, 0, stream);
    k_selfbias<<<gNC, 256, 0, stream>>>(buf0, dinv, bc[1], buf2, NN);
    k_scatter<<<gE32, 256, 0, stream>>>(buf0, dinv, src, dst, buf2, NE);
    k_relu<<<gNC, 256, 0, stream>>>(buf2, NB);
    // conv3: buf2 -> buf1  (h3 = buf1)
    launch_gemm(buf2, Wc[2], buf0, NN, 128, 0, 1.f, nullptr, 0, stream);
    k_selfbias<<<gNC, 256, 0, stream>>>(buf0, dinv, bc[2], buf1, NN);
    k_scatter<<<gE32, 256, 0, stream>>>(buf0, dinv, src, dst, buf1, NE);
    k_relu<<<gNC, 256, 0, stream>>>(buf1, NB);

    // --- a1: per-head GCN K/V (fused across heads), streaming attention pool ---
    k_repack<<<64, 256, 0, stream>>>(Wk1, Bk);
    k_repack<<<64, 256, 0, stream>>>(Wv1, Bv);
    // K = GCN(h3) -> buf2
    launch_gemm(buf1, Bk, buf0, NN, 128, 0, 1.f, nullptr, 0, stream);
    k_selfbias<<<gNC, 256, 0, stream>>>(buf0, dinv, bk1, buf2, NN);
    k_scatter<<<gE32, 256, 0, stream>>>(buf0, dinv, src, dst, buf2, NE);
    // V = GCN(h3) -> buf1 (h3 consumed by this gemm first)
    launch_gemm(buf1, Bv, buf0, NN, 128, 0, 1.f, nullptr, 0, stream);
    k_selfbias<<<gNC, 256, 0, stream>>>(buf0, dinv, bv1, buf1, NN);
    k_scatter<<<gE32, 256, 0, stream>>>(buf0, dinv, src, dst, buf1, NE);
    // q = (seed @ q.T) * DH^-0.5
    launch_gemm(seed1, q1w, wq1, 32, 128, 1, 0.25f, nullptr, 0, stream);
    k_attnpool<<<256, 256, 0, stream>>>(buf2, buf1, wq1, wo1, NN);
    launch_gemm(wo1, Wo1, wp1, 32, 128, 1, 1.f, bo1, 0, stream);
    k_lnadd<<<32, 128, 0, stream>>>(seed1, wp1, lnh1w, lnh1b, wh1);
    launch_gemm(wh1, Wff1, wf1, 32, 128, 1, 1.f, bff1, 0, stream);
    k_lnadd<<<32, 128, 0, stream>>>(wh1, wf1, lnz1w, lnz1b, wz1);

    // --- a2: self-attention over 32 seeds ---
    launch_gemm(wz1, Wi2, wqkv, 32, 384, 1, 1.f, bi2, 0, stream);
    k_selfattn32<<<1, 256, 0, stream>>>(wqkv, wo2);
    launch_gemm(wo2, Wo2, wp2, 32, 128, 1, 1.f, bo2, 0, stream);
    k_lnadd<<<32, 128, 0, stream>>>(wz1, wp2, lnh2w, lnh2b, wh2);
    launch_gemm(wh2, Wff2, wf2, 32, 128, 1, 1.f, bff2, 0, stream);
    k_lnadd<<<32, 128, 0, stream>>>(wh2, wf2, lnz2w, lnz2b, wz2);

    // --- a3: GCN over self-loops == plain linear (deg=2, norm=1/2, x2) ---
    k_repack<<<64, 256, 0, stream>>>(Wk3, Bk);
    k_repack<<<64, 256, 0, stream>>>(Wv3, Bv);
    launch_gemm(wz2, Bk, wk3, 32, 128, 0, 1.f, bk3, 0, stream);
    launch_gemm(wz2, Bv, wv3, 32, 128, 0, 1.f, bv3, 0, stream);
    launch_gemm(seed3, q3w, wq3, 1, 128, 1, 0.25f, nullptr, 0, stream);
    k_attn3<<<1, 128, 0, stream>>>(wk3, wv3, wq3, wo3);
    launch_gemm(wo3, Wo3, wp3, 1, 128, 1, 1.f, bo3, 0, stream);
    k_lnadd<<<1, 128, 0, stream>>>(seed3, wp3, lnh3w, lnh3b, wh3);
    launch_gemm(wh3, Wff3, wf3, 1, 128, 1, 1.f, bff3, 0, stream);
    float* wz3 = wp3;  // reuse
    k_lnadd<<<1, 128, 0, stream>>>(wh3, wf3, lnz3w, lnz3b, wz3);

    // --- head: sigmoid(z @ head_W + head_b) -> d_out [1,32] ---
    launch_gemm(wz3, headW, (float*)d_out, 1, 32, 0, 1.f, headb, 2, stream);
}